// BrainModel_21809843929267
// MI455X (gfx1250) — compile-verified
//
#include <hip/hip_runtime.h>
#include <hip/hip_bf16.h>

// ---------------------------------------------------------------------------
// BrainModel: only the last N_MOTORS(=256) of the N_OUT(=98976) sparse neurons
// feed the output q = sigmoid(pre)[:, -256:] @ w_motor^T + b_motor.
// => compute only those 256 neurons (gather-dot over 32 connections, sigmoid),
//    then a (64x256)@(256x16) f32 GEMM done with V_WMMA_F32_16X16X4_F32.
// ---------------------------------------------------------------------------

#define N_NEURONS 100000
#define N_OUT     98976
#define N_MOTORS  256
#define N_CONN    32
#define BATCH     64
#define N_ACTIONS 16
#define O_BASE    (N_OUT - N_MOTORS)   // 98720: first motor-feeding neuron

typedef __attribute__((ext_vector_type(2))) float v2f;
typedef __attribute__((ext_vector_type(8))) float v8f;

// ---------------------------------------------------------------------------
// Kernel 1: one wave per motor neuron j (256 waves total).
// Lane c preloads idx[o,c] / w[o,c]; the connection loop broadcasts them with
// __shfl (wave32). Each lane accumulates batches (lane) and (lane+32).
// Writes m[b*256 + j] = sigmoid(pre[b, o]) into workspace.
// ---------------------------------------------------------------------------
__global__ __launch_bounds__(128) void brain_sparse_sigmoid(
    const float* __restrict__ x,
    const long long* __restrict__ idx,
    const float* __restrict__ w_sparse,
    const float* __restrict__ b_sparse,
    float* __restrict__ m)
{
    const int lane = threadIdx.x & 31;
    const int wave = threadIdx.x >> 5;
    const int j    = blockIdx.x * 4 + wave;   // 0..255
    const int o    = O_BASE + j;              // neuron index in full net

    // lane c holds connection c of this neuron
    const int   col_l = (int)idx[(size_t)o * N_CONN + lane];
    const float w_l   = w_sparse[(size_t)o * N_CONN + lane];

    const float* xrow0 = x + (size_t)lane        * N_NEURONS;  // batch = lane
    const float* xrow1 = x + (size_t)(lane + 32) * N_NEURONS;  // batch = lane+32

    float acc0 = 0.0f, acc1 = 0.0f;
#pragma unroll
    for (int c = 0; c < N_CONN; ++c) {
        const int   col = __shfl(col_l, c, 32);
        const float w   = __shfl(w_l,   c, 32);
        acc0 = fmaf(xrow0[col], w, acc0);
        acc1 = fmaf(xrow1[col], w, acc1);
    }

    const float bias = b_sparse[o];
    const float s0 = 1.0f / (1.0f + expf(-(acc0 + bias)));
    const float s1 = 1.0f / (1.0f + expf(-(acc1 + bias)));

    m[(size_t)lane        * N_MOTORS + j] = s0;
    m[(size_t)(lane + 32) * N_MOTORS + j] = s1;
}

// ---------------------------------------------------------------------------
// Kernel 2: q(64x16) = m(64x256) @ w_motor^T(256x16) + b_motor, exact f32 via
// V_WMMA_F32_16X16X4_F32. 4 waves; wave t owns batch rows [16t, 16t+16).
//
// VGPR layouts (CDNA5 ISA):
//   A 16x4 f32 (2 VGPRs): lanes 0-15 -> M=lane,    v0=K0 v1=K1
//                         lanes16-31 -> M=lane-16, v0=K2 v1=K3
//   B 4x16  f32 (2 VGPRs): lanes 0-15 -> N=lane,   v0=K0 v1=K1
//                          lanes16-31 -> N=lane-16,v0=K2 v1=K3
//   C/D 16x16 (8 VGPRs): VGPR r: lanes 0-15 -> M=r, N=lane;
//                                lanes16-31 -> M=r+8, N=lane-16
// ---------------------------------------------------------------------------
__global__ __launch_bounds__(128) void brain_motor_wmma(
    const float* __restrict__ m,
    const float* __restrict__ w_motor,   // (16, 256) row-major
    const float* __restrict__ b_motor,   // (16)
    float* __restrict__ q)               // (64, 16) row-major
{
    const int lane = threadIdx.x & 31;
    const int tile = threadIdx.x >> 5;   // 0..3
    const int half = lane >> 4;          // 0 or 1 -> K subpair
    const int l15  = lane & 15;

    const int arow = tile * 16 + l15;    // batch row this lane supplies for A
    const int bcol = l15;                // action col this lane supplies for B

    v8f acc = {};
#pragma unroll 4
    for (int kk = 0; kk < N_MOTORS / 4; ++kk) {
        const int ka = kk * 4 + half * 2;            // even -> 8B aligned
        v2f a = *(const v2f*)(m       + (size_t)arow * N_MOTORS + ka);
        v2f b = *(const v2f*)(w_motor + (size_t)bcol * N_MOTORS + ka);
        // (neg_a, A, neg_b, B, c_mod, C, reuse_a, reuse_b)
        acc = __builtin_amdgcn_wmma_f32_16x16x4_f32(
            false, a, false, b, (short)0, acc, false, false);
    }

    const float bias = b_motor[l15];
#pragma unroll
    for (int r = 0; r < 8; ++r) {
        const int brow = tile * 16 + r + half * 8;
        q[(size_t)brow * N_ACTIONS + l15] = acc[r] + bias;
    }
}

// ---------------------------------------------------------------------------
extern "C" void kernel_launch(void* const* d_in, const int* in_sizes, int n_in,
                              void* d_out, int out_size, void* d_ws, size_t ws_size,
                              hipStream_t stream) {
    const float*     x        = (const float*)d_in[0];
    const long long* idx      = (const long long*)d_in[1];   // int64 indices
    const float*     w_sparse = (const float*)d_in[2];
    const float*     b_sparse = (const float*)d_in[3];
    const float*     w_motor  = (const float*)d_in[4];
    const float*     b_motor  = (const float*)d_in[5];
    float*           q        = (float*)d_out;
    float*           m        = (float*)d_ws;   // 64*256*4 = 64 KB scratch

    brain_sparse_sigmoid<<<N_MOTORS / 4, 128, 0, stream>>>(x, idx, w_sparse, b_sparse, m);
    brain_motor_wmma<<<1, 128, 0, stream>>>(m, w_motor, b_motor, q);
}